// depthconv1d_2680059592713
// MI455X (gfx1250) — compile-verified
//
#include <hip/hip_runtime.h>
#include <stdint.h>

// Depthwise conv1d, B=8, L=16384, C=512, K=31, fp32, SAME padding.
// Bandwidth-bound (AI ~7.7 flop/B). Input staged through LDS with CDNA5
// async global->LDS DMA (global_load_async_to_lds_b128 / s_wait_asynccnt).
// Compute uses a 31-tap register sliding window: 1 LDS load per output row
// instead of 31, so VALU (not LDS banks) is the compute-phase limiter.

#define B_N     8
#define L_N     16384
#define C_N     512
#define K_N     31
#define RAD     15                   // K/2
#define TILE_L  96                   // output rows per block
#define CH      128                  // channels per block
#define ROWS    (TILE_L + K_N - 1)   // 126 staged rows
#define COLS2   (CH / 2)             // 64 float2 columns per row
#define JPT     (TILE_L / 4)         // 24 output rows per thread

__global__ __launch_bounds__(256)
void depthconv1d_gfx1250_kernel(const float* __restrict__ x,
                                const float* __restrict__ w,
                                const float* __restrict__ bias,
                                float* __restrict__ out)
{
    // 126 rows * 128 ch * 4B = 64512 B of LDS
    __shared__ float2 tile[ROWS * COLS2];

    const int tid  = threadIdx.x;
    const int lane = tid & 31;
    const int wv   = tid >> 5;                 // wave id 0..7
    const int lT   = blockIdx.x;               // L tile
    const int cT   = blockIdx.y;               // channel tile 0..3
    const int b    = blockIdx.z;

    const int l0   = lT * TILE_L;              // first output row of tile
    const int c0   = cT * CH;                  // first channel of tile

    // ---------------- stage phase: async DMA global -> LDS ----------------
    // Each wave pulls one 512-byte row per sweep (32 lanes x 16B, coalesced).
    const float* xbase = x + (size_t)b * L_N * C_N + c0;
    for (int r = wv; r < ROWS; r += 8) {
        const int lrow = l0 - RAD + r;
        char* ldsPtr = (char*)tile + r * (CH * 4) + lane * 16;
        if (lrow >= 0 && lrow < L_N) {
            const float* gsrc = xbase + (size_t)lrow * C_N + lane * 4;
            uint32_t ldsAddr = (uint32_t)(size_t)ldsPtr;   // addr[31:0] == LDS offset
            asm volatile("global_load_async_to_lds_b128 %0, %1, off"
                         :: "v"(ldsAddr), "v"(gsrc)
                         : "memory");
        } else {
            float4 z = {0.f, 0.f, 0.f, 0.f};
            *(float4*)ldsPtr = z;              // zero halo outside [0, L)
        }
    }
    asm volatile("s_wait_asynccnt 0" ::: "memory");   // our ASYNCcnt == 0
    __syncthreads();                                  // all waves' rows visible

    // ---------------- compute phase ----------------
    const int col = tid & 63;                  // float2 column (2 channels)
    const int g   = tid >> 6;                  // row group 0..3
    const int c2  = c0 + col * 2;              // absolute channel pair base
    const int jBeg = g * JPT;                  // first local output row

    // 31 taps for this channel pair, held in VGPRs
    float2 wk[K_N];
#pragma unroll
    for (int k = 0; k < K_N; ++k)
        wk[k] = *(const float2*)(w + (size_t)k * C_N + c2);

    const float2 bv = *(const float2*)(bias + c2);

    // Register sliding window over LDS rows: preload rows jBeg..jBeg+29
    // into win[1..30]; each output row then shifts and loads one new row.
    float2 win[K_N];
#pragma unroll
    for (int k = 0; k < K_N - 1; ++k)
        win[k + 1] = tile[(jBeg + k) * COLS2 + col];

#pragma unroll
    for (int j = 0; j < JPT; ++j) {
#pragma unroll
        for (int k = 0; k < K_N - 1; ++k)      // pure SSA renaming after unroll
            win[k] = win[k + 1];
        win[K_N - 1] = tile[(jBeg + j + K_N - 1) * COLS2 + col];

        float2 acc = bv;
#pragma unroll
        for (int k = 0; k < K_N; ++k) {
            acc.x = fmaf(win[k].x, wk[k].x, acc.x);
            acc.y = fmaf(win[k].y, wk[k].y, acc.y);
        }

        const int l = l0 + jBeg + j;
        if (l < L_N)
            *(float2*)(out + ((size_t)b * L_N + l) * C_N + c2) = acc;
    }
}

extern "C" void kernel_launch(void* const* d_in, const int* in_sizes, int n_in,
                              void* d_out, int out_size, void* d_ws, size_t ws_size,
                              hipStream_t stream)
{
    const float* x    = (const float*)d_in[0];   // [8, 16384, 512]
    const float* w    = (const float*)d_in[1];   // [31, 512, 1]
    const float* bias = (const float*)d_in[2];   // [512]
    float* out        = (float*)d_out;           // [8, 16384, 512]

    dim3 grid((L_N + TILE_L - 1) / TILE_L,       // 171 L-tiles
              C_N / CH,                          // 4 channel tiles
              B_N);                              // 8 batches
    depthconv1d_gfx1250_kernel<<<grid, 256, 0, stream>>>(x, w, bias, out);
}